// CoordsSelect_79585743995276
// MI455X (gfx1250) — compile-verified
//
#include <hip/hip_runtime.h>
#include <cmath>
#include <cstdint>

#define TPB 256

// grid = (chunksPerBatch, B). Stage this block's coords chunk (TPB atoms * 3
// floats) into LDS via the gfx1250 async global->LDS path (ASYNCcnt), then
// gather F features per atom with one big load clause.
template <int F>
__global__ void __launch_bounds__(TPB)
coords_select_kernel(const float* __restrict__ volume,
                     const float* __restrict__ coords,
                     const int*   __restrict__ num_atoms,
                     float*       __restrict__ out,
                     int A, int D)
{
    __shared__ float sc[TPB * 3];

    const int t  = threadIdx.x;
    const int b  = blockIdx.y;
    const int a0 = blockIdx.x * TPB;
    const int D3 = D * D * D;

    // ---- async stage of coords[b, 3*a0 .. 3*(a0+TPB)) into LDS -------------
    // INST_OFFSET advances both the global and LDS address (ISA
    // CDNA5 ISA), so one LDS-base VGPR + one voffset VGPR +
    // immediate offsets cover all 3*TPB dwords.
    int nd = 3 * (A - a0);                 // valid dwords in this chunk
    if (nd > 3 * TPB) nd = 3 * TPB;

    const float* gbase   = coords + (size_t)b * 3u * (size_t)A + (size_t)a0 * 3u;
    unsigned     ldsAddr = (unsigned)(uintptr_t)(void*)&sc[t]; // low 32b = LDS offset
    unsigned     voff    = (unsigned)t * 4u;

    if (t < nd)
        asm volatile("global_load_async_to_lds_b32 %0, %1, %2 offset:0"
                     :: "v"(ldsAddr), "v"(voff), "s"(gbase) : "memory");
    if (TPB + t < nd)
        asm volatile("global_load_async_to_lds_b32 %0, %1, %2 offset:1024"
                     :: "v"(ldsAddr), "v"(voff), "s"(gbase) : "memory");
    if (2 * TPB + t < nd)
        asm volatile("global_load_async_to_lds_b32 %0, %1, %2 offset:2048"
                     :: "v"(ldsAddr), "v"(voff), "s"(gbase) : "memory");
    asm volatile("s_wait_asynccnt 0x0" ::: "memory");
    __syncthreads();

    const int a = a0 + t;
    if (a >= A) return;

    // ---- voxel index ------------------------------------------------------
    const float res = 120.0f / (float)D;   // reference: RES = BOX_ANG / D
    float x = sc[3 * t + 0];
    float y = sc[3 * t + 1];
    float z = sc[3 * t + 2];
    int ix = (int)floorf(x / res);
    int iy = (int)floorf(y / res);
    int iz = (int)floorf(z / res);
    ix = ix < 0 ? 0 : (ix > D - 1 ? D - 1 : ix);
    iy = iy < 0 ? 0 : (iy > D - 1 ? D - 1 : iy);
    iz = iz < 0 ? 0 : (iz > D - 1 ? D - 1 : iz);
    const int flat = ix * D * D + iy * D + iz;

    const float* vb = volume + (size_t)b * (size_t)F * (size_t)D3 + (size_t)flat;
    float*       ob = out    + (size_t)b * (size_t)F * (size_t)A  + (size_t)a;

    float v[F];
    if (a < num_atoms[b]) {
#pragma unroll
        for (int f = 0; f < F; ++f)        // one 16-deep global_load clause
            v[f] = vb[(size_t)f * D3];
    } else {
#pragma unroll
        for (int f = 0; f < F; ++f)        // masked atoms -> zeros (d_out is poisoned)
            v[f] = 0.0f;
    }
#pragma unroll
    for (int f = 0; f < F; ++f)            // coalesced store clause
        ob[(size_t)f * A] = v[f];
}

// Runtime-F fallback (same structure, un-templated loop)
__global__ void __launch_bounds__(TPB)
coords_select_kernel_gen(const float* __restrict__ volume,
                         const float* __restrict__ coords,
                         const int*   __restrict__ num_atoms,
                         float*       __restrict__ out,
                         int A, int D, int F)
{
    __shared__ float sc[TPB * 3];
    const int t  = threadIdx.x;
    const int b  = blockIdx.y;
    const int a0 = blockIdx.x * TPB;
    const int D3 = D * D * D;

    int nd = 3 * (A - a0);
    if (nd > 3 * TPB) nd = 3 * TPB;
    const float* gbase   = coords + (size_t)b * 3u * (size_t)A + (size_t)a0 * 3u;
    unsigned     ldsAddr = (unsigned)(uintptr_t)(void*)&sc[t];
    unsigned     voff    = (unsigned)t * 4u;
    if (t < nd)
        asm volatile("global_load_async_to_lds_b32 %0, %1, %2 offset:0"
                     :: "v"(ldsAddr), "v"(voff), "s"(gbase) : "memory");
    if (TPB + t < nd)
        asm volatile("global_load_async_to_lds_b32 %0, %1, %2 offset:1024"
                     :: "v"(ldsAddr), "v"(voff), "s"(gbase) : "memory");
    if (2 * TPB + t < nd)
        asm volatile("global_load_async_to_lds_b32 %0, %1, %2 offset:2048"
                     :: "v"(ldsAddr), "v"(voff), "s"(gbase) : "memory");
    asm volatile("s_wait_asynccnt 0x0" ::: "memory");
    __syncthreads();

    const int a = a0 + t;
    if (a >= A) return;

    const float res = 120.0f / (float)D;
    int ix = (int)floorf(sc[3 * t + 0] / res);
    int iy = (int)floorf(sc[3 * t + 1] / res);
    int iz = (int)floorf(sc[3 * t + 2] / res);
    ix = ix < 0 ? 0 : (ix > D - 1 ? D - 1 : ix);
    iy = iy < 0 ? 0 : (iy > D - 1 ? D - 1 : iy);
    iz = iz < 0 ? 0 : (iz > D - 1 ? D - 1 : iz);
    const int flat = ix * D * D + iy * D + iz;

    const float* vb = volume + (size_t)b * (size_t)F * (size_t)D3 + (size_t)flat;
    float*       ob = out    + (size_t)b * (size_t)F * (size_t)A  + (size_t)a;
    const bool active = (a < num_atoms[b]);
    for (int f = 0; f < F; ++f)
        ob[(size_t)f * A] = active ? vb[(size_t)f * D3] : 0.0f;
}

extern "C" void kernel_launch(void* const* d_in, const int* in_sizes, int n_in,
                              void* d_out, int out_size, void* d_ws, size_t ws_size,
                              hipStream_t stream) {
    const float* volume    = (const float*)d_in[0];
    const float* coords    = (const float*)d_in[1];
    const int*   num_atoms = (const int*)d_in[2];
    float*       out       = (float*)d_out;

    // Derive shapes: in_sizes = { B*F*D^3, B*3*A, B }, out_size = B*F*A
    const int B = in_sizes[2];
    const int A = in_sizes[1] / (3 * B);
    const int F = out_size / (B * A);
    const double d3 = (double)in_sizes[0] / ((double)B * (double)F);
    const int D = (int)llround(cbrt(d3));

    const int chunksPerBatch = (A + TPB - 1) / TPB;
    dim3 grid(chunksPerBatch, B);          // no runtime div/mod in the kernel
    dim3 block(TPB);

    if (F == 16) {
        coords_select_kernel<16><<<grid, block, 0, stream>>>(
            volume, coords, num_atoms, out, A, D);
    } else {
        coords_select_kernel_gen<<<grid, block, 0, stream>>>(
            volume, coords, num_atoms, out, A, D, F);
    }
}